// SpatialTransformer_4337916969678
// MI455X (gfx1250) — compile-verified
//
#include <hip/hip_runtime.h>

// Spatial transformer (dense 3D trilinear warp), fp32.
// vol: [B,D,H,W,C] f32, trf: [B,D,H,W,3] f32 -> out: [B,D,H,W,C] f32
// Memory-bound gather: ~229 MB min HBM traffic -> ~10us floor @ 23.3 TB/s.
// Strategy:
//  - 1 block per (b,z,y) row, 160 lanes = x: index decomposition is uniform
//    -> SALU, killing the v_mul_hi/v_ashr div-by-160 chains seen in round 1.
//  - NT hints on streaming trf (b96 load) / out (b64 store) so vol (65.5 MB)
//    stays resident in the 192 MB L2 while trf+out (164 MB) stream through.
//  - 8 x b64 gathers for the C=2 channel pairs (near-coalesced: lanes differ
//    by x +- a few voxels).
//  - global_prefetch_b8 ~96KB ahead on the trf stream.

typedef float v2f __attribute__((ext_vector_type(2)));

constexpr int B = 2, D = 160, H = 160, W = 160, C = 2;
constexpr int HW    = H * W;          // 25,600
constexpr int DHW   = D * HW;         // 4,096,000
constexpr int NVOX  = B * DHW;        // 8,192,000
constexpr int NROWS = B * D * H;      // 51,200 blocks
constexpr int PF_DIST = 8192;         // voxels ahead => ~96KB ahead in trf

__global__ __launch_bounds__(W)
void SpatialTransformer_warp3d(const float* __restrict__ vol,
                               const float* __restrict__ trf,
                               float* __restrict__ out) {
    // Uniform (per-block) row decomposition -> scalar ALU.
    const unsigned row = blockIdx.x;        // (b*D + z)*H + y
    const unsigned y   = row % H;
    const unsigned t   = row / H;
    const unsigned z   = t % D;
    const unsigned b   = t / D;

    const unsigned x = threadIdx.x;         // lane = x coordinate
    const unsigned i = row * W + x;         // flat voxel index

    // Prefetch the streaming displacement field ahead (global_prefetch_b8).
    unsigned ip = i + PF_DIST;
    if (ip >= NVOX) ip = NVOX - 1;
    __builtin_prefetch(trf + 3ull * ip, 0, 1);

    // Read-once displacement: non-temporal b96 load.
    const size_t ti = 3ull * i;
    const float sz = __builtin_nontemporal_load(trf + ti + 0);
    const float sy = __builtin_nontemporal_load(trf + ti + 1);
    const float sx = __builtin_nontemporal_load(trf + ti + 2);

    // Absolute sample location (unclipped), exactly as reference.
    const float lz = (float)z + sz;
    const float ly = (float)y + sy;
    const float lx = (float)x + sx;

    // Clipped floor corner, then clipped ceil corner.
    const float z0c = fminf(fmaxf(floorf(lz), 0.f), (float)(D - 1));
    const float y0c = fminf(fmaxf(floorf(ly), 0.f), (float)(H - 1));
    const float x0c = fminf(fmaxf(floorf(lx), 0.f), (float)(W - 1));
    const float z1c = fminf(z0c + 1.f, (float)(D - 1));
    const float y1c = fminf(y0c + 1.f, (float)(H - 1));
    const float x1c = fminf(x0c + 1.f, (float)(W - 1));

    // Weights from clipped ceil vs UNclipped loc (reference semantics):
    // d1 (pairs with floor corner) = loc1c - loc ; d0 (ceil corner) = 1 - d1.
    const float wz0 = z1c - lz, wz1 = 1.f - wz0;
    const float wy0 = y1c - ly, wy1 = 1.f - wy0;
    const float wx0 = x1c - lx, wx1 = 1.f - wx0;

    const int iz0 = (int)z0c, iz1 = (int)z1c;
    const int iy0 = (int)y0c, iy1 = (int)y1c;
    const int ix0 = (int)x0c, ix1 = (int)x1c;

    const float* __restrict__ vb = vol + (size_t)b * (size_t)DHW * C;
    const int z0o = iz0 * HW, z1o = iz1 * HW;
    const int y0o = iy0 * W,  y1o = iy1 * W;

    const float w00 = wz0 * wy0, w01 = wz0 * wy1;
    const float w10 = wz1 * wy0, w11 = wz1 * wy1;

    float a0 = 0.f, a1 = 0.f;
#define CORNER(zo, yo, xi, w)                                                   \
    {                                                                           \
        const v2f v = *(const v2f*)(vb + (size_t)(unsigned)((zo) + (yo) + (xi)) * C); \
        a0 = fmaf((w), v.x, a0);                                                \
        a1 = fmaf((w), v.y, a1);                                                \
    }
    // Corner order matches itertools.product([0,1],repeat=3): (z,y,x) bits.
    CORNER(z0o, y0o, ix0, w00 * wx0);
    CORNER(z0o, y0o, ix1, w00 * wx1);
    CORNER(z0o, y1o, ix0, w01 * wx0);
    CORNER(z0o, y1o, ix1, w01 * wx1);
    CORNER(z1o, y0o, ix0, w10 * wx0);
    CORNER(z1o, y0o, ix1, w10 * wx1);
    CORNER(z1o, y1o, ix0, w11 * wx0);
    CORNER(z1o, y1o, ix1, w11 * wx1);
#undef CORNER

    // Write-once result: non-temporal b64 store.
    v2f res;
    res.x = a0;
    res.y = a1;
    __builtin_nontemporal_store(res, (v2f*)(out + (size_t)i * C));
}

extern "C" void kernel_launch(void* const* d_in, const int* in_sizes, int n_in,
                              void* d_out, int out_size, void* d_ws, size_t ws_size,
                              hipStream_t stream) {
    const float* vol = (const float*)d_in[0];
    const float* trf = (const float*)d_in[1];
    float* out = (float*)d_out;
    (void)in_sizes; (void)n_in; (void)out_size; (void)d_ws; (void)ws_size;

    dim3 block(W);        // 160 lanes = one row of x
    dim3 grid(NROWS);     // 51,200 rows (b,z,y)
    SpatialTransformer_warp3d<<<grid, block, 0, stream>>>(vol, trf, out);
}